// GATBlock_77421080477767
// MI455X (gfx1250) — compile-verified
//
#include <hip/hip_runtime.h>
#include <hip/hip_bf16.h>
#include <math.h>

#define DIM 128
#define LDSW 136                 // padded LDS row stride (floats): keeps half-wave
                                 // B-fragment reads in disjoint banks (2*136 % 64 == 16)
#define NEG_SLOPE 0.2f
#define NUM_GRAPHS 64

typedef __attribute__((ext_vector_type(2))) float v2f;
typedef __attribute__((ext_vector_type(8))) float v8f;

// ---------------------------------------------------------------------------
// H[N x 128] = X[N x 128] @ W[128 x 128] via V_WMMA_F32_16X16X4_F32, with the
// per-row attention dots fused into the epilogue:
//   as[i] = H[i,:].a_src ,  ad[i] = H[i,:].a_dst
//
// Block = 256 threads (8 waves). W is staged once per block into LDS (68KB,
// padded stride). Each wave owns one 16-row tile and computes the FULL 16x128
// stripe with 8 accumulators, so one A fragment feeds 8 WMMAs per k-step:
//   per k-step: 2 global_load_b32 (A) + 16 ds_load_b32 (B) + 8 v_wmma.
// X is read from global exactly once. N == 50000 == 3125*16 (no row guards
// inside a tile).
//
// Fragment layouts per CDNA5 ISA 7.12.2:
//   A (16x4):  lanes 0-15 -> M=lane, {K=k,k+1}; lanes 16-31 -> {k+2,k+3}
//   B (4x16):  lanes 0-15 -> N=lane, {K=k,k+1}; lanes 16-31 -> {k+2,k+3}
//   C (16x16): vgpr r -> M = r (lanes 0-15) / r+8 (lanes 16-31), N = lane&15
// ---------------------------------------------------------------------------
__global__ __launch_bounds__(256) void gemm_dots_wmma_f32(
    const float* __restrict__ X, const float* __restrict__ W,
    const float* __restrict__ a_src, const float* __restrict__ a_dst,
    float* __restrict__ H, float* __restrict__ as_, float* __restrict__ ad_,
    int N)
{
  __shared__ float Wl[DIM * LDSW];     // 128*136*4 = 69632 bytes

  // ---- stage W into LDS (float4, coalesced; each thread copies 16 vec4) ----
  {
    const float4* W4 = (const float4*)W;
#pragma unroll
    for (int i = 0; i < 16; ++i) {
      const int t4  = threadIdx.x + i * 256;   // 0..4095
      const int row = t4 >> 5;                 // t4*4 / 128
      const int col = (t4 & 31) * 4;
      *(float4*)&Wl[row * LDSW + col] = W4[t4];
    }
  }
  __syncthreads();

  const int wave = threadIdx.x >> 5;
  const int lane = threadIdx.x & 31;
  const int half = lane >> 4;                  // 0 or 1
  const int l15  = lane & 15;

  const int tile = blockIdx.x * 8 + wave;      // 16-row tile index
  if (tile >= N / 16) return;
  const int row0 = tile * 16;

  v8f acc[8];
  const v8f vz = {0.f, 0.f, 0.f, 0.f, 0.f, 0.f, 0.f, 0.f};
#pragma unroll
  for (int j = 0; j < 8; ++j) acc[j] = vz;

  const float* Xrow = X + (size_t)(row0 + l15) * DIM;

#pragma unroll 4
  for (int k = 0; k < DIM; k += 4) {
    const int ak = k + 2 * half;
    v2f a;
    a.x = Xrow[ak];
    a.y = Xrow[ak + 1];
    const float* b0 = &Wl[ak * LDSW + l15];
    const float* b1 = &Wl[(ak + 1) * LDSW + l15];
#pragma unroll
    for (int j = 0; j < 8; ++j) {
      v2f b;
      b.x = b0[16 * j];
      b.y = b1[16 * j];
      acc[j] = __builtin_amdgcn_wmma_f32_16x16x4_f32(
          false, a, false, b, (short)0, acc[j], false, false);
    }
  }

  // ---- store H stripe ----
#pragma unroll
  for (int j = 0; j < 8; ++j) {
#pragma unroll
    for (int r = 0; r < 8; ++r) {
      const int rr = row0 + r + 8 * half;
      H[(size_t)rr * DIM + 16 * j + l15] = acc[j][r];
    }
  }

  // ---- fused attention dots: rows live in this wave's registers ----
  float asv[8], adv[8];
#pragma unroll
  for (int j = 0; j < 8; ++j) {
    asv[j] = a_src[16 * j + l15];
    adv[j] = a_dst[16 * j + l15];
  }
#pragma unroll
  for (int r = 0; r < 8; ++r) {
    float p = 0.f, q = 0.f;
#pragma unroll
    for (int j = 0; j < 8; ++j) {
      p += acc[j][r] * asv[j];
      q += acc[j][r] * adv[j];
    }
    // reduce across the 16 lanes of each half-wave (masks < 16 stay in-half)
#pragma unroll
    for (int msk = 8; msk >= 1; msk >>= 1) {
      p += __shfl_xor(p, msk, 32);
      q += __shfl_xor(q, msk, 32);
    }
    if (l15 == 0) {
      const int rr = row0 + r + 8 * half;
      as_[rr] = p;
      ad_[rr] = q;
    }
  }
}

// float atomic max via signed/unsigned integer ordering trick (init to -inf)
__device__ __forceinline__ void atomicMaxF(float* addr, float val) {
  if (val >= 0.f) atomicMax((int*)addr, __float_as_int(val));
  else            atomicMin((unsigned int*)addr, __float_as_uint(val));
}

// Pass 1: per-destination running max of leaky-relu'd logits.
__global__ void edge_max(const int* __restrict__ ei, int E, int N,
                         const float* __restrict__ as_, const float* __restrict__ ad_,
                         float* __restrict__ m)
{
  const int e = blockIdx.x * blockDim.x + threadIdx.x;
  const int Ef = E + N;
  if (e >= Ef) return;
  const int s = (e < E) ? ei[e]     : (e - E);
  const int d = (e < E) ? ei[E + e] : (e - E);
  float l = as_[s] + ad_[d];
  l = (l > 0.f) ? l : NEG_SLOPE * l;
  atomicMaxF(&m[d], l);
}

// Pass 2: exp(logit - max[dst]) cached per edge, denominator accumulated.
__global__ void edge_expsum(const int* __restrict__ ei, int E, int N,
                            const float* __restrict__ as_, const float* __restrict__ ad_,
                            const float* __restrict__ m, float* __restrict__ ex,
                            float* __restrict__ denom)
{
  const int e = blockIdx.x * blockDim.x + threadIdx.x;
  const int Ef = E + N;
  if (e >= Ef) return;
  const int s = (e < E) ? ei[e]     : (e - E);
  const int d = (e < E) ? ei[E + e] : (e - E);
  float l = as_[s] + ad_[d];
  l = (l > 0.f) ? l : NEG_SLOPE * l;
  const float v = expf(l - m[d]);
  ex[e] = v;
  atomicAdd(&denom[d], v);
}

// Pass 3: out[dst,:] += (ex[e]/denom[dst]) * H[src,:]. One wave per edge;
// each lane owns 4 consecutive features (float4 load + 4 f32 atomic adds).
__global__ __launch_bounds__(256) void edge_aggregate(
    const int* __restrict__ ei, int E, int N,
    const float* __restrict__ ex, const float* __restrict__ denom,
    const float* __restrict__ H, float* __restrict__ out)
{
  const int wave = threadIdx.x >> 5;
  const int lane = threadIdx.x & 31;
  const int e = blockIdx.x * 8 + wave;
  const int Ef = E + N;
  if (e >= Ef) return;
  const int s = (e < E) ? ei[e]     : (e - E);
  const int d = (e < E) ? ei[E + e] : (e - E);
  const float alpha = ex[e] / denom[d];
  const float4 h = ((const float4*)(H + (size_t)s * DIM))[lane];
  float* o = out + (size_t)d * DIM + lane * 4;
  atomicAdd(o + 0, alpha * h.x);
  atomicAdd(o + 1, alpha * h.y);
  atomicAdd(o + 2, alpha * h.z);
  atomicAdd(o + 3, alpha * h.w);
}

// Bias add (+ optional ReLU) in place, then accumulate pooled per-graph sums.
__global__ __launch_bounds__(256) void bias_act_pool(
    float* __restrict__ out, const float* __restrict__ b,
    const int* __restrict__ batch, float* __restrict__ pooled,
    int N, int do_relu)
{
  const int wave = threadIdx.x >> 5;
  const int lane = threadIdx.x & 31;
  const int node = blockIdx.x * 8 + wave;
  if (node >= N) return;
  const int g = batch[node];
  float4* o4 = (float4*)(out + (size_t)node * DIM);
  float4 v = o4[lane];
  const float4 bb = ((const float4*)b)[lane];
  v.x += bb.x; v.y += bb.y; v.z += bb.z; v.w += bb.w;
  if (do_relu) {
    v.x = fmaxf(v.x, 0.f); v.y = fmaxf(v.y, 0.f);
    v.z = fmaxf(v.z, 0.f); v.w = fmaxf(v.w, 0.f);
  }
  o4[lane] = v;
  float* p = pooled + (size_t)g * DIM + lane * 4;
  atomicAdd(p + 0, v.x);
  atomicAdd(p + 1, v.y);
  atomicAdd(p + 2, v.z);
  atomicAdd(p + 3, v.w);
}

__global__ void count_nodes(const int* __restrict__ batch, float* __restrict__ cnt, int N) {
  const int n = blockIdx.x * blockDim.x + threadIdx.x;
  if (n < N) atomicAdd(&cnt[batch[n]], 1.0f);
}

__global__ void pool_div(const float* __restrict__ pooled, const float* __restrict__ cnt,
                         float* __restrict__ emb) {
  const int t = blockIdx.x * blockDim.x + threadIdx.x;
  if (t >= NUM_GRAPHS * DIM) return;
  emb[t] = pooled[t] / fmaxf(cnt[t / DIM], 1.0f);
}

__global__ void fill_f32(float* __restrict__ p, float v, int n) {
  const int i = blockIdx.x * blockDim.x + threadIdx.x;
  if (i < n) p[i] = v;
}

// ---------------------------------------------------------------------------
extern "C" void kernel_launch(void* const* d_in, const int* in_sizes, int n_in,
                              void* d_out, int out_size, void* d_ws, size_t ws_size,
                              hipStream_t stream) {
  const float* x      = (const float*)d_in[0];
  const int*   ei     = (const int*)  d_in[1];
  // d_in[2] = edge_weight (unused by the reference)
  const int*   batch  = (const int*)  d_in[3];
  const float* W0     = (const float*)d_in[4];
  const float* asrc0  = (const float*)d_in[5];
  const float* adst0  = (const float*)d_in[6];
  const float* b0     = (const float*)d_in[7];
  const float* W1     = (const float*)d_in[8];
  const float* asrc1  = (const float*)d_in[9];
  const float* adst1  = (const float*)d_in[10];
  const float* b1     = (const float*)d_in[11];

  const int N  = in_sizes[0] / DIM;   // 50000
  const int E  = in_sizes[1] / 2;     // 600000
  const int Ef = E + N;               // edges incl. self loops

  // ---- workspace carve-up (256B aligned) ----
  char* ws = (char*)d_ws;
  size_t off = 0;
  auto carve = [&](size_t bytes) -> char* {
    char* p = ws + off;
    off = (off + bytes + 255) & ~(size_t)255;
    return p;
  };
  float* H      = (float*)carve((size_t)N * DIM * sizeof(float));
  float* OUT    = (float*)carve((size_t)N * DIM * sizeof(float));
  float* as_    = (float*)carve((size_t)N * sizeof(float));
  float* ad_    = (float*)carve((size_t)N * sizeof(float));
  float* m      = (float*)carve((size_t)N * sizeof(float));
  float* denom  = (float*)carve((size_t)N * sizeof(float));
  float* ex     = (float*)carve((size_t)Ef * sizeof(float));
  float* pooled = (float*)carve((size_t)NUM_GRAPHS * DIM * sizeof(float));
  float* cnt    = (float*)carve((size_t)NUM_GRAPHS * sizeof(float));

  float* emb0 = (float*)d_out;
  float* emb1 = (float*)d_out + NUM_GRAPHS * DIM;

  const int TB = 256;
  const dim3 blk(TB);
  auto blocks1 = [&](int n) { return dim3((n + TB - 1) / TB); };

  const int tiles = N / 16;                      // 3125
  const dim3 gemm_grid((tiles + 7) / 8);         // 8 waves (tiles) per block

  // per-graph node counts (shared by both layers)
  fill_f32<<<blocks1(NUM_GRAPHS), blk, 0, stream>>>(cnt, 0.f, NUM_GRAPHS);
  count_nodes<<<blocks1(N), blk, 0, stream>>>(batch, cnt, N);

  const float* Xl = x;
  for (int layer = 0; layer < 2; ++layer) {
    const float* W    = layer ? W1    : W0;
    const float* asrc = layer ? asrc1 : asrc0;
    const float* adst = layer ? adst1 : adst0;
    const float* b    = layer ? b1    : b0;
    float* emb        = layer ? emb1  : emb0;

    // 1) H = Xl @ W (WMMA f32, W staged in LDS) + fused attention dots
    gemm_dots_wmma_f32<<<gemm_grid, blk, 0, stream>>>(Xl, W, asrc, adst,
                                                      H, as_, ad_, N);
    // 2) reset per-layer accumulators (OUT is safe to clear: gemm consumed Xl)
    fill_f32<<<blocks1(N), blk, 0, stream>>>(m, -INFINITY, N);
    fill_f32<<<blocks1(N), blk, 0, stream>>>(denom, 0.f, N);
    fill_f32<<<blocks1(N * DIM), blk, 0, stream>>>(OUT, 0.f, N * DIM);
    fill_f32<<<blocks1(NUM_GRAPHS * DIM), blk, 0, stream>>>(pooled, 0.f, NUM_GRAPHS * DIM);
    // 3) segment softmax over edges
    edge_max<<<blocks1(Ef), blk, 0, stream>>>(ei, E, N, as_, ad_, m);
    edge_expsum<<<blocks1(Ef), blk, 0, stream>>>(ei, E, N, as_, ad_, m, ex, denom);
    // 4) weighted aggregation: one wave per edge
    edge_aggregate<<<dim3((Ef + 7) / 8), blk, 0, stream>>>(ei, E, N, ex, denom, H, OUT);
    // 5) bias (+ReLU on layer 0) and pooled sums
    bias_act_pool<<<dim3((N + 7) / 8), blk, 0, stream>>>(OUT, b, batch, pooled, N,
                                                         layer == 0 ? 1 : 0);
    // 6) mean pool
    pool_div<<<blocks1(NUM_GRAPHS * DIM), blk, 0, stream>>>(pooled, cnt, emb);

    Xl = OUT;  // layer-1 input is the (ReLU'd, biased) layer-0 output
  }
}